// MultiHopHeteroGCNLayer_18425409699949
// MI455X (gfx1250) — compile-verified
//
#include <hip/hip_runtime.h>

typedef __attribute__((ext_vector_type(2))) float v2f;
typedef __attribute__((ext_vector_type(8))) float v8f;

// ---------------------------------------------------------------------------
// Problem constants (from reference)
// ---------------------------------------------------------------------------
#define NRELS 10
#define IN_F 512
#define OUT_F 128
#define LDS_STRIDE 516   // 512 + 4 pad: lane m hits bank (4*m)%64 -> conflict-free
#define TILE_M 32        // rows per block; each wave does two 16x16 tiles sharing B

// ---------------------------------------------------------------------------
// Utility kernels
// ---------------------------------------------------------------------------
__global__ void zero_f32_kernel(float* __restrict__ p, int n) {
    int i = blockIdx.x * blockDim.x + threadIdx.x;
    if (i < n) p[i] = 0.0f;
}

__global__ void count_deg_kernel(const int* __restrict__ src, const int* __restrict__ dst,
                                 float* __restrict__ cnt_out, float* __restrict__ cnt_in,
                                 int n_edges) {
    int i = blockIdx.x * blockDim.x + threadIdx.x;
    if (i < n_edges) {
        atomicAdd(&cnt_out[src[i]], 1.0f);
        atomicAdd(&cnt_in[dst[i]], 1.0f);
    }
}

__global__ void rsqrt_clamp_kernel(float* __restrict__ p, int n) {
    int i = blockIdx.x * blockDim.x + threadIdx.x;
    if (i < n) {
        float v = p[i];
        v = v < 1.0f ? 1.0f : v;
        p[i] = rsqrtf(v);
    }
}

// ---------------------------------------------------------------------------
// WMMA GEMM: Y[n_rows x 128] = X[n_rows x 512] @ W[512 x 128], fp32.
// Block: 256 threads = 8 waves, computing a 32 x 128 output tile.
// Wave w owns N-subtile n0 = 16*w and TWO M-subtiles (rows 0-15, 16-31 of the
// block tile), reusing each B fragment for two V_WMMA_F32_16X16X4_F32 ops.
// n_rows is always a multiple of 32 here (4000/8000/16000).
// ---------------------------------------------------------------------------
__global__ __launch_bounds__(256)
void gemm_xw_wmma_kernel(const float* __restrict__ X, const float* __restrict__ W,
                         float* __restrict__ Y) {
    __shared__ float lds_x[TILE_M * LDS_STRIDE];

    const int tile_m = blockIdx.x * TILE_M;

    // Cooperatively stage 32x512 X tile into LDS (float4, coalesced).
    const float4* __restrict__ Xv = reinterpret_cast<const float4*>(X + (size_t)tile_m * IN_F);
#pragma unroll
    for (int i = 0; i < 16; ++i) {
        int idx  = threadIdx.x + i * 256;     // 0..4095 float4 elements
        int row  = idx >> 7;                  // 0..31
        int col4 = idx & 127;                 // 0..127
        float4 v = Xv[idx];
        *reinterpret_cast<float4*>(&lds_x[row * LDS_STRIDE + col4 * 4]) = v;
    }
    __syncthreads();

    const int lane = threadIdx.x & 31;
    const int wave = threadIdx.x >> 5;
    const int half = lane >> 4;   // 0: lanes 0-15, 1: lanes 16-31
    const int m    = lane & 15;
    const int n0   = wave * 16;

    v8f c0 = {};                  // rows tile_m + 0..15
    v8f c1 = {};                  // rows tile_m + 16..31
#pragma unroll 4
    for (int k0 = 0; k0 < IN_F; k0 += 4) {
        // B fragment (4x16 f32): v0 = K rows {k0, k0+2}, v1 = {k0+1, k0+3}; N = n0+m
        const float* bp = W + (size_t)(k0 + half * 2) * OUT_F + n0 + m;
        v2f b;
        b.x = bp[0];
        b.y = bp[OUT_F];
        // A fragments (16x4 f32) for the two M-subtiles
        const float* ap0 = &lds_x[m * LDS_STRIDE + k0 + half * 2];
        const float* ap1 = &lds_x[(m + 16) * LDS_STRIDE + k0 + half * 2];
        v2f a0, a1;
        a0.x = ap0[0]; a0.y = ap0[1];
        a1.x = ap1[0]; a1.y = ap1[1];
        c0 = __builtin_amdgcn_wmma_f32_16x16x4_f32(
                false, a0, false, b, (short)0, c0, false, false);
        c1 = __builtin_amdgcn_wmma_f32_16x16x4_f32(
                false, a1, false, b, (short)0, c1, false, false);
    }

    // D layout: c[i] -> M = i + half*8, N = n0 + m
    float* yp0 = Y + (size_t)tile_m * OUT_F + n0 + m;
    float* yp1 = yp0 + (size_t)16 * OUT_F;
#pragma unroll
    for (int i = 0; i < 8; ++i) {
        yp0[(size_t)(i + half * 8) * OUT_F] = c0[i];
        yp1[(size_t)(i + half * 8) * OUT_F] = c1[i];
    }
}

// ---------------------------------------------------------------------------
// Edge scatter: acc[dst] += Y[src] * s_out[src] * s_in[dst]
// One wave (32 threads) per edge, each thread handles 4 consecutive columns.
// ---------------------------------------------------------------------------
__global__ void scatter_edges_kernel(const float* __restrict__ Y,
                                     const int* __restrict__ src, const int* __restrict__ dst,
                                     const float* __restrict__ s_out, const float* __restrict__ s_in,
                                     float* __restrict__ acc, int n_edges) {
    int t = blockIdx.x * blockDim.x + threadIdx.x;
    int e = t >> 5;
    if (e >= n_edges) return;
    int q = (t & 31) * 4;                 // column chunk 0..124
    int s = src[e];
    int d = dst[e];
    float scale = s_out[s] * s_in[d];
    float4 v = *reinterpret_cast<const float4*>(Y + (size_t)s * OUT_F + q);
    float* o = acc + (size_t)d * OUT_F + q;
    atomicAdd(o + 0, v.x * scale);
    atomicAdd(o + 1, v.y * scale);
    atomicAdd(o + 2, v.z * scale);
    atomicAdd(o + 3, v.w * scale);
}

// ---------------------------------------------------------------------------
// Finalize: out = relu(out + sum of biases of relations targeting this type)
// ---------------------------------------------------------------------------
__global__ void finalize_kernel(float* __restrict__ out,
                                const float* __restrict__ b0, const float* __restrict__ b1,
                                const float* __restrict__ b2, const float* __restrict__ b3,
                                int n_elems) {
    int i = blockIdx.x * blockDim.x + threadIdx.x;
    if (i >= n_elems) return;
    int c = i & (OUT_F - 1);
    float bias = 0.0f;
    if (b0) bias += b0[c];
    if (b1) bias += b1[c];
    if (b2) bias += b2[c];
    if (b3) bias += b3[c];
    float v = out[i] + bias;
    out[i] = v > 0.0f ? v : 0.0f;
}

// ---------------------------------------------------------------------------
// Host launcher
// ---------------------------------------------------------------------------
extern "C" void kernel_launch(void* const* d_in, const int* in_sizes, int n_in,
                              void* d_out, int out_size, void* d_ws, size_t ws_size,
                              hipStream_t stream) {
    (void)in_sizes; (void)n_in; (void)out_size; (void)ws_size;

    // Relation tables, in RELS order:
    // dd, pp, dp, pd, ddis, disd, dse, sed, pdis, disp
    static const int NSRC [NRELS] = {8000, 16000,  8000, 16000,  8000,  4000,  8000,  4000, 16000,  4000};
    static const int NDST [NRELS] = {8000, 16000, 16000,  8000,  4000,  8000,  4000,  8000,  4000, 16000};
    static const int NEDGE[NRELS] = {200000, 800000, 400000, 400000, 100000, 100000, 100000, 100000, 150000, 150000};
    static const int STYPE[NRELS] = {0, 1, 0, 1, 0, 2, 0, 3, 1, 2};  // source node type
    static const int DTYPE[NRELS] = {0, 1, 1, 0, 2, 0, 3, 0, 2, 1};  // dest node type
    static const int NTYPE[4]    = {8000, 16000, 4000, 4000};        // drug, protein, disease, sideeffect
    static const int TROW [4]    = {0, 8000, 24000, 28000};          // row offset of each type in d_out

    // Input pointers
    const float* h_type[4];
    for (int t = 0; t < 4; ++t) h_type[t] = (const float*)d_in[t];
    const float* Wr[NRELS];
    const float* br[NRELS];
    const int*   srcr[NRELS];
    const int*   dstr[NRELS];
    for (int r = 0; r < NRELS; ++r) {
        Wr[r]   = (const float*)d_in[4 + 2 * r];
        br[r]   = (const float*)d_in[5 + 2 * r];
        srcr[r] = (const int*)d_in[24 + 2 * r];
        dstr[r] = (const int*)d_in[25 + 2 * r];
    }
    float* out = (float*)d_out;

    // Workspace layout (floats):
    //  [0 .. )           Y_r buffers  (sum NSRC[r] * 128)
    //  [yTot .. )        per-relation s_out (NSRC[r]) then s_in (NDST[r])
    float* wsf = (float*)d_ws;
    size_t yoff[NRELS];
    size_t cur = 0;
    for (int r = 0; r < NRELS; ++r) { yoff[r] = cur; cur += (size_t)NSRC[r] * OUT_F; }
    size_t soff_out[NRELS], soff_in[NRELS];
    size_t scaleBase = cur;
    for (int r = 0; r < NRELS; ++r) { soff_out[r] = cur; cur += NSRC[r];
                                      soff_in [r] = cur; cur += NDST[r]; }
    int nScales = (int)(cur - scaleBase);

    const int nOut = 32000 * OUT_F;

    // 1) Zero output accumulator + degree counters
    zero_f32_kernel<<<(nOut + 255) / 256, 256, 0, stream>>>(out, nOut);
    zero_f32_kernel<<<(nScales + 255) / 256, 256, 0, stream>>>(wsf + scaleBase, nScales);

    // 2) Degree counts (atomic) per relation
    for (int r = 0; r < NRELS; ++r) {
        count_deg_kernel<<<(NEDGE[r] + 255) / 256, 256, 0, stream>>>(
            srcr[r], dstr[r], wsf + soff_out[r], wsf + soff_in[r], NEDGE[r]);
    }

    // 3) counts -> rsqrt(clamp(count, 1))
    rsqrt_clamp_kernel<<<(nScales + 255) / 256, 256, 0, stream>>>(wsf + scaleBase, nScales);

    // 4) WMMA GEMMs: Y_r = X_src @ W_r  (row scaling folded into scatter)
    for (int r = 0; r < NRELS; ++r) {
        gemm_xw_wmma_kernel<<<NSRC[r] / TILE_M, 256, 0, stream>>>(
            h_type[STYPE[r]], Wr[r], wsf + yoff[r]);
    }

    // 5) Edge scatter: out[type(d)][dst] += Y_r[src] * s_out[src] * s_in[dst]
    for (int r = 0; r < NRELS; ++r) {
        long long threads = (long long)NEDGE[r] * 32;
        scatter_edges_kernel<<<(unsigned)((threads + 255) / 256), 256, 0, stream>>>(
            wsf + yoff[r], srcr[r], dstr[r],
            wsf + soff_out[r], wsf + soff_in[r],
            out + (size_t)TROW[DTYPE[r]] * OUT_F, NEDGE[r]);
    }

    // 6) Bias + ReLU per node type
    // drug <- dd(0), pd(3), disd(5), sed(7); protein <- pp(1), dp(2), disp(9);
    // disease <- ddis(4), pdis(8); sideeffect <- dse(6)
    {
        int ne;
        ne = NTYPE[0] * OUT_F;
        finalize_kernel<<<(ne + 255) / 256, 256, 0, stream>>>(
            out + (size_t)TROW[0] * OUT_F, br[0], br[3], br[5], br[7], ne);
        ne = NTYPE[1] * OUT_F;
        finalize_kernel<<<(ne + 255) / 256, 256, 0, stream>>>(
            out + (size_t)TROW[1] * OUT_F, br[1], br[2], br[9], nullptr, ne);
        ne = NTYPE[2] * OUT_F;
        finalize_kernel<<<(ne + 255) / 256, 256, 0, stream>>>(
            out + (size_t)TROW[2] * OUT_F, br[4], br[8], nullptr, nullptr, ne);
        ne = NTYPE[3] * OUT_F;
        finalize_kernel<<<(ne + 255) / 256, 256, 0, stream>>>(
            out + (size_t)TROW[3] * OUT_F, br[6], nullptr, nullptr, nullptr, ne);
    }
}